// GAT_8624294330678
// MI455X (gfx1250) — compile-verified
//
#include <hip/hip_runtime.h>

#define NNODES 50000
#define NEDGES 800000
#define FEATS  128
#define HEADS  4
#define HID    32
#define NEG_SLOPE 0.2f

typedef float v2f __attribute__((ext_vector_type(2)));
typedef float v8f __attribute__((ext_vector_type(8)));

__device__ __forceinline__ float lrelu(float v) { return v > 0.f ? v : NEG_SLOPE * v; }

// Order-preserving float -> uint mapping so we can use fast integer atomicMax.
__device__ __forceinline__ unsigned fkey(float f) {
    unsigned u = __float_as_uint(f);
    return (u & 0x80000000u) ? ~u : (u | 0x80000000u);
}
__device__ __forceinline__ float funkey(unsigned u) {
    return __uint_as_float((u & 0x80000000u) ? (u & 0x7fffffffu) : ~u);
}

// ---------------------------------------------------------------------------
// feat = X @ W  (X: n x 128 fp32, W: 128 x 128 fp32, row-major)
// One wave computes a 16-row x 128-col slab with V_WMMA_F32_16X16X4_F32.
//
// W is staged in LDS K-pair-major as float2: entry (p, c) = {W[2p][c], W[2p+1][c]},
// p = 0..63, c = 0..127, with a bank swizzle (c + (p&1)*16) & 127 so the two
// lane halves (pair p vs pair p+1) hit disjoint bank windows.  Each B fragment
// is then a single aligned ds_load_b64 into an even VGPR pair (no repacking).
//
// A fragment (16x4 f32): lanes 0-15 hold K=0,1 of row M=lane; lanes 16-31 hold
// K=2,3 of row M=lane-16.  C/D: vgpr r -> row r (lanes 0-15) / r+8 (lanes 16-31).
// ---------------------------------------------------------------------------
__global__ __launch_bounds__(256) void k_gemm(const float* __restrict__ X,
                                              const float* __restrict__ W,
                                              float* __restrict__ feat, int n) {
    __shared__ float sW[FEATS * FEATS];  // 64 KB, viewed as float2[64][128]
    {
        float2* sW2 = (float2*)sW;
        for (int idx = threadIdx.x; idx < 64 * 64; idx += blockDim.x) {
            int p = idx >> 6;
            int c = (idx & 63) * 2;
            const float2 r0 = *(const float2*)(W + (size_t)(2 * p) * FEATS + c);
            const float2 r1 = *(const float2*)(W + (size_t)(2 * p + 1) * FEATS + c);
            int sw = (p & 1) << 4;
            sW2[p * 128 + ((c + sw) & 127)]     = make_float2(r0.x, r1.x);
            sW2[p * 128 + ((c + 1 + sw) & 127)] = make_float2(r0.y, r1.y);
        }
    }
    __syncthreads();

    const int wave = threadIdx.x >> 5;
    const int lane = threadIdx.x & 31;
    const int slab = blockIdx.x * 8 + wave;
    const int nSlabs = (n + 15) >> 4;
    if (slab >= nSlabs) return;  // uniform per wave: EXEC stays all-ones for WMMA

    const int nodeBase = slab << 4;
    const int half = lane >> 4;   // 0: K-pair p, 1: K-pair p+1
    const int lrow = lane & 15;
    int arow = nodeBase + lrow;
    if (arow >= n) arow = n - 1;
    const float* xrow = X + (size_t)arow * FEATS;
    const float2* sW2 = (const float2*)sW;
    const int csw = lrow + (half << 4);  // swizzled column base for this lane

    v8f acc[8];
    const v8f zero = {0.f, 0.f, 0.f, 0.f, 0.f, 0.f, 0.f, 0.f};
#pragma unroll
    for (int t = 0; t < 8; ++t) acc[t] = zero;

#pragma unroll 4
    for (int k0 = 0; k0 < FEATS; k0 += 4) {
        const int p = (k0 >> 1) + half;          // this lane's K-pair
        const float2 av = *(const float2*)(xrow + 2 * p);
        v2f a;
        a.x = av.x;
        a.y = av.y;
#pragma unroll
        for (int t = 0; t < 8; ++t) {
            const float2 bv = sW2[p * 128 + ((t * 16 + csw) & 127)];  // ds_load_b64
            v2f b;
            b.x = bv.x;
            b.y = bv.y;
            acc[t] = __builtin_amdgcn_wmma_f32_16x16x4_f32(
                false, a, false, b, (short)0, acc[t], false, false);
        }
    }

#pragma unroll
    for (int t = 0; t < 8; ++t) {
#pragma unroll
        for (int r = 0; r < 8; ++r) {
            int row = nodeBase + r + 8 * half;
            if (row < n) feat[(size_t)row * FEATS + t * 16 + lrow] = acc[t][r];
        }
    }
}

// el[n,h] = sum_d feat[n,h,d]*al[h,d];  er analogous.  One wave per node.
__global__ __launch_bounds__(256) void k_elr(const float* __restrict__ feat,
                                             const float* __restrict__ al,
                                             const float* __restrict__ ar,
                                             float* __restrict__ el,
                                             float* __restrict__ er, int n) {
    int node = blockIdx.x * 8 + (threadIdx.x >> 5);
    if (node >= n) return;
    int lane = threadIdx.x & 31;
    const float* f = feat + (size_t)node * FEATS;
    float accl[HEADS], accr[HEADS];
#pragma unroll
    for (int h = 0; h < HEADS; ++h) {
        float v = f[h * HID + lane];
        accl[h] = v * al[h * HID + lane];
        accr[h] = v * ar[h * HID + lane];
    }
#pragma unroll
    for (int off = 16; off > 0; off >>= 1) {
#pragma unroll
        for (int h = 0; h < HEADS; ++h) {
            accl[h] += __shfl_xor(accl[h], off, 32);
            accr[h] += __shfl_xor(accr[h], off, 32);
        }
    }
    if (lane == 0) {
#pragma unroll
        for (int h = 0; h < HEADS; ++h) {
            el[node * HEADS + h] = accl[h];
            er[node * HEADS + h] = accr[h];
        }
    }
}

// Pass 1 of edge softmax: per-dst per-head running max (integer atomicMax on keys).
__global__ __launch_bounds__(256) void k_edge_max(const int* __restrict__ src,
                                                  const int* __restrict__ dst,
                                                  const float* __restrict__ el,
                                                  const float* __restrict__ er,
                                                  unsigned* __restrict__ emax, int e) {
    int i = blockIdx.x * blockDim.x + threadIdx.x;
    if (i >= e) return;
    int s = src[i], d = dst[i];
    const float4 l4 = *(const float4*)(el + s * 4);
    const float4 r4 = *(const float4*)(er + d * 4);
    atomicMax(emax + d * 4 + 0, fkey(lrelu(l4.x + r4.x)));
    atomicMax(emax + d * 4 + 1, fkey(lrelu(l4.y + r4.y)));
    atomicMax(emax + d * 4 + 2, fkey(lrelu(l4.z + r4.z)));
    atomicMax(emax + d * 4 + 3, fkey(lrelu(l4.w + r4.w)));
}

// Pass 2: ex = exp(e - max), accumulate softmax denominators.
__global__ __launch_bounds__(256) void k_edge_exp(const int* __restrict__ src,
                                                  const int* __restrict__ dst,
                                                  const float* __restrict__ el,
                                                  const float* __restrict__ er,
                                                  const unsigned* __restrict__ emax,
                                                  float* __restrict__ denom,
                                                  float* __restrict__ ex, int e) {
    int i = blockIdx.x * blockDim.x + threadIdx.x;
    if (i >= e) return;
    int s = src[i], d = dst[i];
    const float4 l4 = *(const float4*)(el + s * 4);
    const float4 r4 = *(const float4*)(er + d * 4);
    float4 xv;
    xv.x = __expf(lrelu(l4.x + r4.x) - funkey(emax[d * 4 + 0]));
    xv.y = __expf(lrelu(l4.y + r4.y) - funkey(emax[d * 4 + 1]));
    xv.z = __expf(lrelu(l4.z + r4.z) - funkey(emax[d * 4 + 2]));
    xv.w = __expf(lrelu(l4.w + r4.w) - funkey(emax[d * 4 + 3]));
    *(float4*)(ex + (size_t)i * 4) = xv;
    atomicAdd(denom + d * 4 + 0, xv.x);
    atomicAdd(denom + d * 4 + 1, xv.y);
    atomicAdd(denom + d * 4 + 2, xv.z);
    atomicAdd(denom + d * 4 + 3, xv.w);
}

// One wave per edge: gather feat[src] (float4/lane), scale by alpha, scatter-add.
__global__ __launch_bounds__(256) void k_aggregate(const int* __restrict__ src,
                                                   const int* __restrict__ dst,
                                                   const float* __restrict__ feat,
                                                   const float* __restrict__ ex,
                                                   const float* __restrict__ denom,
                                                   float* __restrict__ agg, int e) {
    int edge = blockIdx.x * 8 + (threadIdx.x >> 5);
    if (edge >= e) return;
    int lane = threadIdx.x & 31;
    int s = src[edge], d = dst[edge];
    int h = lane >> 3;  // dims 4*lane..4*lane+3 all belong to head lane/8
    float alpha = ex[(size_t)edge * 4 + h] / denom[d * 4 + h];
    const float4 f = *(const float4*)(feat + (size_t)s * FEATS + lane * 4);
    float* o = agg + (size_t)d * FEATS + lane * 4;
    atomicAdd(o + 0, f.x * alpha);
    atomicAdd(o + 1, f.y * alpha);
    atomicAdd(o + 2, f.z * alpha);
    atomicAdd(o + 3, f.w * alpha);
}

__global__ __launch_bounds__(256) void k_elu_bias(float* __restrict__ h,
                                                  const float* __restrict__ b, int total) {
    int i = blockIdx.x * blockDim.x + threadIdx.x;
    if (i >= total) return;
    float v = h[i] + b[i & (FEATS - 1)];
    h[i] = v > 0.f ? v : __expf(v) - 1.f;
}

// out[n,o] = mean_h (agg[n,h,o] + b2[h,o])
__global__ __launch_bounds__(256) void k_mean(const float* __restrict__ agg,
                                              const float* __restrict__ b2,
                                              float* __restrict__ out, int n) {
    int i = blockIdx.x * blockDim.x + threadIdx.x;
    if (i >= n * 32) return;
    int node = i >> 5;
    int o = i & 31;
    float s = 0.f;
#pragma unroll
    for (int h = 0; h < HEADS; ++h) s += agg[(size_t)node * FEATS + h * HID + o] + b2[h * HID + o];
    out[i] = s * 0.25f;
}

extern "C" void kernel_launch(void* const* d_in, const int* in_sizes, int n_in,
                              void* d_out, int out_size, void* d_ws, size_t ws_size,
                              hipStream_t stream) {
    const int N = NNODES, E = NEDGES;
    const float* x = (const float*)d_in[0];
    const int* src = (const int*)d_in[1];
    const int* dst = (const int*)d_in[2];
    const float* W[3]  = {(const float*)d_in[3], (const float*)d_in[7],  (const float*)d_in[11]};
    const float* al[3] = {(const float*)d_in[4], (const float*)d_in[8],  (const float*)d_in[12]};
    const float* ar[3] = {(const float*)d_in[5], (const float*)d_in[9],  (const float*)d_in[13]};
    const float* b[3]  = {(const float*)d_in[6], (const float*)d_in[10], (const float*)d_in[14]};

    float* feat = (float*)d_ws;                      // N*128
    float* hbuf = feat + (size_t)N * FEATS;          // N*128 (agg / activation, reused)
    float* el   = hbuf + (size_t)N * FEATS;          // N*4
    float* er   = el + (size_t)N * HEADS;            // N*4
    unsigned* emax = (unsigned*)(er + (size_t)N * HEADS);  // N*4
    float* denom = (float*)(emax + (size_t)N * HEADS);     // N*4
    float* ex    = denom + (size_t)N * HEADS;              // E*4

    const int nSlabs = (N + 15) / 16;
    const int gemmBlocks = (nSlabs + 7) / 8;

    const float* layerIn = x;
    for (int l = 0; l < 3; ++l) {
        k_gemm<<<gemmBlocks, 256, 0, stream>>>(layerIn, W[l], feat, N);
        k_elr<<<(N + 7) / 8, 256, 0, stream>>>(feat, al[l], ar[l], el, er, N);
        // zero-init scratch (graph-capturable memset nodes); hbuf is free after gemm read it
        hipMemsetAsync(emax, 0, (size_t)N * HEADS * sizeof(unsigned), stream);
        hipMemsetAsync(denom, 0, (size_t)N * HEADS * sizeof(float), stream);
        hipMemsetAsync(hbuf, 0, (size_t)N * FEATS * sizeof(float), stream);
        k_edge_max<<<(E + 255) / 256, 256, 0, stream>>>(src, dst, el, er, emax, E);
        k_edge_exp<<<(E + 255) / 256, 256, 0, stream>>>(src, dst, el, er, emax, denom, ex, E);
        k_aggregate<<<(E + 7) / 8, 256, 0, stream>>>(src, dst, feat, ex, denom, hbuf, E);
        if (l < 2) {
            k_elu_bias<<<(N * FEATS + 255) / 256, 256, 0, stream>>>(hbuf, b[l], N * FEATS);
            layerIn = hbuf;
        } else {
            k_mean<<<(N * 32 + 255) / 256, 256, 0, stream>>>(hbuf, b[l], (float*)d_out, N);
        }
    }
}